// GeometricSuperpositionSearch_45715631898881
// MI455X (gfx1250) — compile-verified
//
#include <hip/hip_runtime.h>
#include <hip/hip_bf16.h>

// ---------------------------------------------------------------------------
// GeometricSuperpositionSearch, folded form.
//
// Key algebraic identity: new_state[b,n,:] = state[b,n,:] @ Cop[b]  where
//   Cop[b][j][l] = a[b][j^l] * cayley_sign(i=j^l, j),
//   a[b][i]      = sum_k softmax_w[b,k] * instructions[b,k,i].
// This removes the [B,K,N,16] intermediate entirely. The problem becomes
// HBM-bound: read 33.5MB twice + write 33.5MB  => ~4.3us at 23.3 TB/s.
// The per-tile [16x16]x[16x16] product uses V_WMMA_F32_16X16X4_F32 (fp32,
// matching reference precision exactly).
// ---------------------------------------------------------------------------

#define GB 8          // batch
#define GN 65536      // cells
#define GD 16         // multivector dim
#define GK 8          // templates
#define GM 8          // rule memory slots
#define GHID 64
#define GEPS 1e-12f

typedef float v2f __attribute__((ext_vector_type(2)));
typedef float v8f __attribute__((ext_vector_type(8)));

// ---------------- phase 0: zero the summary accumulator ----------------
__global__ void zero_ws_kernel(float* p, int n) {
    int i = blockIdx.x * blockDim.x + threadIdx.x;
    if (i < n) p[i] = 0.0f;
}

// ---------------- phase 1: cpu_state sum over N (float4 + LDS + atomics) ---
__global__ __launch_bounds__(256)
void summary_kernel(const float* __restrict__ state, float* __restrict__ ws_sum) {
    const int b = blockIdx.y;
    const float4* p = (const float4*)(state + (size_t)b * GN * GD);
    const int total4 = GN * GD / 4;          // 262144 float4's per batch
    const int stride = gridDim.x * blockDim.x; // multiple of 4 -> quad stable
    float4 s = make_float4(0.f, 0.f, 0.f, 0.f);
    for (int idx = blockIdx.x * blockDim.x + threadIdx.x; idx < total4; idx += stride) {
        float4 v = p[idx];
        s.x += v.x; s.y += v.y; s.z += v.z; s.w += v.w;
    }
    __shared__ float4 sdata[256];
    sdata[threadIdx.x] = s;
    __syncthreads();
    // threads with equal (t & 3) share the same d-quad; fold down to t<4
    for (int off = 128; off >= 4; off >>= 1) {
        if (threadIdx.x < off) {
            float4 o = sdata[threadIdx.x + off];
            sdata[threadIdx.x].x += o.x; sdata[threadIdx.x].y += o.y;
            sdata[threadIdx.x].z += o.z; sdata[threadIdx.x].w += o.w;
        }
        __syncthreads();
    }
    if (threadIdx.x < 4) {
        float4 r = sdata[threadIdx.x];
        float* dst = ws_sum + b * GD + threadIdx.x * 4;
        atomicAdd(dst + 0, r.x); atomicAdd(dst + 1, r.y);
        atomicAdd(dst + 2, r.z); atomicAdd(dst + 3, r.w);
    }
}

// ---------------- threefry2x32, exactly as JAX (key = PRNGKey(42) = [0,42]) -
__device__ __forceinline__ unsigned rotl32(unsigned x, int r) {
    return (x << r) | (x >> (32 - r));
}
__device__ float jax_uniform64(int idx) {
    // counts = iota(64) split into halves -> pairs (i, i+32)
    const unsigned k0 = 0u, k1 = 42u, k2 = 0x1BD11BDAu ^ k0 ^ k1;
    const unsigned ks[3] = {k0, k1, k2};
    const int R[8] = {13, 15, 26, 6, 17, 29, 16, 24};
    int p = (idx < 32) ? idx : idx - 32;
    unsigned x0 = (unsigned)p + k0;
    unsigned x1 = (unsigned)(p + 32) + k1;
    for (int g = 0; g < 5; ++g) {
        #pragma unroll
        for (int r = 0; r < 4; ++r) {
            x0 += x1;
            x1 = rotl32(x1, R[(g & 1) * 4 + r]);
            x1 ^= x0;
        }
        x0 += ks[(g + 1) % 3];
        x1 += ks[(g + 2) % 3] + (unsigned)(g + 1);
    }
    unsigned bits = (idx < 32) ? x0 : x1;
    float f = __uint_as_float((bits >> 9) | 0x3f800000u) - 1.0f; // [0,1)
    const float lo = 1e-6f, hi = 1.0f - 1e-6f;
    return fmaxf(lo, f * (hi - lo) + lo);
}

// ---------------- phase 2: control path + fold into Cop[b][16][16] ---------
__global__ __launch_bounds__(256)
void control_kernel(const float* __restrict__ ws_sum,
                    const float* __restrict__ ctrl,
                    const float* __restrict__ rule_mem,
                    const float* __restrict__ templates,
                    const float* __restrict__ W1, const float* __restrict__ b1,
                    const float* __restrict__ W2, const float* __restrict__ b2,
                    const float* __restrict__ Wr, const float* __restrict__ br,
                    const float* __restrict__ log_temp,
                    float* __restrict__ ws_cop) {
    __shared__ float a_sh[GB][GD];
    const int t = threadIdx.x;
    if (t < GB) {
        const int b = t;
        float summ[GD];
        #pragma unroll
        for (int d = 0; d < GD; ++d) summ[d] = ws_sum[b * GD + d] * (1.0f / (float)GN);
        // per-grade norms of the mean multivector
        float gn[5] = {GEPS, GEPS, GEPS, GEPS, GEPS};
        #pragma unroll
        for (int d = 0; d < GD; ++d) gn[__popc(d)] += summ[d] * summ[d];
        float si[9];
        #pragma unroll
        for (int g = 0; g < 5; ++g) si[g] = sqrtf(gn[g]);
        #pragma unroll
        for (int c = 0; c < 4; ++c) si[5 + c] = ctrl[b * 4 + c];
        // MLP 9 -> 64 -> 8
        float h[GHID];
        for (int j = 0; j < GHID; ++j) {
            float acc = b1[j];
            #pragma unroll
            for (int i = 0; i < 9; ++i) acc += si[i] * W1[i * GHID + j];
            h[j] = fmaxf(acc, 0.0f);
        }
        float scores[GK];
        #pragma unroll
        for (int k = 0; k < GK; ++k) {
            float acc = b2[k];
            for (int j = 0; j < GHID; ++j) acc += h[j] * W2[j * GK + k];
            scores[k] = acc;
        }
        // rule summary (mean over M)
        float rs[GD];
        #pragma unroll
        for (int d = 0; d < GD; ++d) {
            float acc = 0.f;
            #pragma unroll
            for (int m = 0; m < GM; ++m) acc += rule_mem[((size_t)b * GM + m) * GD + d];
            rs[d] = acc * (1.0f / (float)GM);
        }
        // gumbel-softmax weights (threefry matches jax.random.key(42))
        float tau = expf(log_temp[0]);
        tau = fminf(fmaxf(tau, 0.1f), 5.0f);
        float w[GK]; float mx = -1e30f;
        #pragma unroll
        for (int k = 0; k < GK; ++k) {
            float u = jax_uniform64(b * GK + k);
            float g = -logf(-logf(u));
            w[k] = (scores[k] + g) / tau;
            mx = fmaxf(mx, w[k]);
        }
        float sum = 0.f;
        #pragma unroll
        for (int k = 0; k < GK; ++k) { w[k] = expf(w[k] - mx); sum += w[k]; }
        const float inv = 1.0f / sum;
        // fold: a[i] = sum_k w_k * scores_k * (templates[k][i] + (rs@Wr+br)[k][i])
        #pragma unroll
        for (int i = 0; i < GD; ++i) {
            float acc = 0.f;
            #pragma unroll
            for (int k = 0; k < GK; ++k) {
                float rm = br[k * GD + i];
                #pragma unroll
                for (int d = 0; d < GD; ++d) rm += rs[d] * Wr[d * (GK * GD) + k * GD + i];
                acc += (w[k] * inv) * scores[k] * (templates[k * GD + i] + rm);
            }
            a_sh[b][i] = acc;
        }
    }
    __syncthreads();
    // Cop[b][j][l] = a[b][j^l] * cayley_sign(i=j^l, j); Cl(3,0,1): e0^2 = 0
    for (int e = t; e < GB * GD * GD; e += blockDim.x) {
        const int b = e >> 8, j = (e >> 4) & 15, l = e & 15;
        const int i = j ^ l;
        float s;
        if (i & j & 1) {
            s = 0.0f;                       // shared e0 -> metric contraction 0
        } else {
            int sgn = 0;
            #pragma unroll
            for (int bit = 0; bit < 4; ++bit)
                if ((j >> bit) & 1) sgn += __popc(i >> (bit + 1));
            s = (sgn & 1) ? -1.0f : 1.0f;
        }
        ws_cop[e] = a_sh[b][i] * s;
    }
}

// ---------------- phase 3: out[b] = state[b] (Nx16) @ Cop[b] via WMMA ------
// V_WMMA_F32_16X16X4_F32: A 16x4, B 4x16, C/D 16x16, all fp32.
// A layout (32-bit A 16x4): lanes 0-15 -> row M=lane, v0=K0 v1=K1;
//                           lanes 16-31 -> row M=lane-16, v0=K2 v1=K3.
// B layout (transpose-symmetric): lanes 0-15 -> col N=lane, v0=K0 v1=K1;
//                                 lanes 16-31 -> col N=lane-16, v0=K2 v1=K3.
// D layout: VGPR r: lanes 0-15 -> (M=r, N=lane); lanes 16-31 -> (M=8+r, N=lane-16).
#define TILES_PER_WAVE 4
__global__ __launch_bounds__(256)
void apply_op_kernel(const float* __restrict__ state,
                     const float* __restrict__ cop,
                     float* __restrict__ out) {
    const int lane = threadIdx.x & 31;
    const int wave = threadIdx.x >> 5;
    const int wave_global = blockIdx.x * (blockDim.x >> 5) + wave;
    const int waves_per_batch = (GN / 16) / TILES_PER_WAVE;   // 1024
    const int b  = wave_global / waves_per_batch;
    const int wi = wave_global % waves_per_batch;
    const int row16 = lane & 15;      // row within tile (A) / column (B, D)
    const int khalf = lane >> 4;      // 0: K pair {0,1}; 1: K pair {2,3}

    // B-matrix fragments for this batch: chunk c covers j = 4c..4c+3
    const float* copb = cop + (size_t)b * 256;
    v2f bm[4];
    #pragma unroll
    for (int c = 0; c < 4; ++c) {
        const int j0 = 4 * c + 2 * khalf;
        bm[c].x = copb[j0 * 16 + row16];
        bm[c].y = copb[(j0 + 1) * 16 + row16];
    }

    const size_t base_row = (size_t)b * GN + (size_t)wi * (16 * TILES_PER_WAVE);
    for (int tt = 0; tt < TILES_PER_WAVE; ++tt) {
        const float* arow = state + (base_row + (size_t)tt * 16) * GD;
        v8f acc = {};
        #pragma unroll
        for (int c = 0; c < 4; ++c) {
            const float* pa = arow + row16 * GD + c * 4 + 2 * khalf;
            v2f am; am.x = pa[0]; am.y = pa[1];
            acc = __builtin_amdgcn_wmma_f32_16x16x4_f32(
                false, am, false, bm[c], (short)0, acc, false, false);
        }
        float* orow = out + (base_row + (size_t)tt * 16) * GD;
        #pragma unroll
        for (int r = 0; r < 8; ++r) {
            orow[(r + 8 * khalf) * GD + row16] = acc[r];
        }
    }
}

extern "C" void kernel_launch(void* const* d_in, const int* in_sizes, int n_in,
                              void* d_out, int out_size, void* d_ws, size_t ws_size,
                              hipStream_t stream) {
    const float* cpu_state   = (const float*)d_in[0];
    const float* ctrl_cursor = (const float*)d_in[1];
    const float* rule_memory = (const float*)d_in[2];
    const float* templates   = (const float*)d_in[3];
    const float* W1 = (const float*)d_in[4];
    const float* b1 = (const float*)d_in[5];
    const float* W2 = (const float*)d_in[6];
    const float* b2 = (const float*)d_in[7];
    const float* Wr = (const float*)d_in[8];
    const float* br = (const float*)d_in[9];
    const float* log_temperature = (const float*)d_in[10];
    float* out = (float*)d_out;

    float* ws_sum = (float*)d_ws;            // 128 floats (B*16 accumulators)
    float* ws_cop = (float*)d_ws + 256;      // 2048 floats (B*16*16 operators)

    zero_ws_kernel<<<1, 128, 0, stream>>>(ws_sum, GB * GD);
    summary_kernel<<<dim3(128, GB), 256, 0, stream>>>(cpu_state, ws_sum);
    control_kernel<<<1, 256, 0, stream>>>(ws_sum, ctrl_cursor, rule_memory,
                                          templates, W1, b1, W2, b2, Wr, br,
                                          log_temperature, ws_cop);
    // 8192 waves total, 4 tiles each -> 32768 tiles = B * N/16
    apply_op_kernel<<<1024, 256, 0, stream>>>(cpu_state, ws_cop, out);
}